// LogNeuralJastrowSlater_16020228014421
// MI455X (gfx1250) — compile-verified
//
#include <hip/hip_runtime.h>
#include <math.h>

typedef __attribute__((ext_vector_type(2))) float v2f;
typedef __attribute__((ext_vector_type(8))) float v8f;

#define N_ORB 32
#define N_F   16
#define HID   64
#define PI_F  3.14159265358979323846f

// One wave handles a tile of 16 batch rows:
//  - occupation bitmasks via coalesced loads + ballot
//  - Jastrow GEMM (16x32 @ 32x64) via v_wmma_f32_16x16x4_f32 (exact fp32)
//  - two 16x16 LU decompositions at a time (one per 16-lane half), 8 pairs total
__global__ __launch_bounds__(128) void njs_kernel(
    const float* __restrict__ nmat,
    const float* __restrict__ Mmat,
    const float* __restrict__ Wmat,
    const float* __restrict__ bvec,
    float* __restrict__ out,
    int nTiles)
{
    __shared__ float ldsJ[4 * 16];   // per-wave slice of 16 J values

    const int lane         = threadIdx.x & 31;
    const int waveInBlock  = threadIdx.x >> 5;
    const int wavesPerBlk  = blockDim.x >> 5;
    const int globalWave   = blockIdx.x * wavesPerBlk + waveInBlock;
    const int totalWaves   = gridDim.x * wavesPerBlk;
    const int i            = lane & 15;   // row-in-half
    const int hi           = lane >> 4;   // which 16-lane half
    const int hbase        = hi << 4;

    for (int tile = globalWave; tile < nTiles; tile += totalWaves) {
        const int rowBase = tile * 16;

        // ---- 1) occupation bitmasks (coalesced, single pass over n) ----
        // lanes i and i+16 both end up holding mask of batch row rowBase+i
        unsigned mask = 0u;
        #pragma unroll
        for (int j = 0; j < 16; ++j) {
            float v = nmat[(rowBase + j) * N_ORB + lane];
            unsigned bal = (unsigned)__ballot(v > 0.5f);
            if (i == j) mask = bal;
        }

        // ---- 2) WMMA A-fragments for f32 16x16x4 (A = n tile, 16x32) ----
        // lane<16:  VGPR0 = A[M=i, k0+0], VGPR1 = A[M=i, k0+1]
        // lane>=16: VGPR0 = A[M=i, k0+2], VGPR1 = A[M=i, k0+3]
        v2f afr[8];
        #pragma unroll
        for (int ks = 0; ks < 8; ++ks) {
            int k = ks * 4 + 2 * hi;
            afr[ks].x = (float)((mask >> k) & 1u);
            afr[ks].y = (float)((mask >> (k + 1)) & 1u);
        }

        // ---- 3) Jastrow: D = n@W over 4 column tiles of 16, tanh + row sum ----
        float s[8];
        #pragma unroll
        for (int r = 0; r < 8; ++r) s[r] = 0.0f;

        #pragma unroll
        for (int c = 0; c < 4; ++c) {
            v8f acc = {};
            #pragma unroll
            for (int ks = 0; ks < 8; ++ks) {
                int krow = ks * 4 + 2 * hi;
                v2f bfr;
                bfr.x = Wmat[(krow    ) * HID + c * 16 + i];
                bfr.y = Wmat[(krow + 1) * HID + c * 16 + i];
                acc = __builtin_amdgcn_wmma_f32_16x16x4_f32(
                        false, afr[ks], false, bfr, (short)0, acc, false, false);
            }
            float bb = bvec[c * 16 + i];
            #pragma unroll
            for (int r = 0; r < 8; ++r) s[r] += tanhf(acc[r] + bb);
        }
        // reduce over the 16 lanes of each half: s[r] -> J[row r + 8*hi]
        #pragma unroll
        for (int d = 8; d >= 1; d >>= 1) {
            #pragma unroll
            for (int r = 0; r < 8; ++r) s[r] += __shfl_xor(s[r], d, 32);
        }
        // park J in LDS so the (rolled) determinant loop can index it dynamically
        if (i == 0) {
            #pragma unroll
            for (int r = 0; r < 8; ++r)
                ldsJ[waveInBlock * 16 + hi * 8 + r] = s[r];
        }
        asm volatile("s_wait_dscnt 0" ::: "memory");  // same-wave LDS ordering

        // ---- 4) slogdet of 16x16 gathered matrices, two per pass ----
        #pragma unroll 1
        for (int t = 0; t < 8; ++t) {
            // half hi handles batch row rowBase + 2t + hi
            unsigned mq = (unsigned)__shfl((int)mask, 2 * t + hi, 32);

            // R[i] = index of (i+1)-th set bit (ascending occupied orbitals)
            unsigned mm = mq;
            #pragma unroll
            for (int cbit = 0; cbit < 15; ++cbit)
                if (cbit < i) mm &= (mm - 1u);
            int Ri = (int)__builtin_ctz(mm);

            // lane i holds row i of A = M[R] : 16 floats (4x float4, 64B aligned)
            float a[16];
            {
                const float4* Mr = (const float4*)(Mmat + Ri * N_F);
                float4 q0 = Mr[0], q1 = Mr[1], q2 = Mr[2], q3 = Mr[3];
                a[0]=q0.x;  a[1]=q0.y;  a[2]=q0.z;  a[3]=q0.w;
                a[4]=q1.x;  a[5]=q1.y;  a[6]=q1.z;  a[7]=q1.w;
                a[8]=q2.x;  a[9]=q2.y;  a[10]=q2.z; a[11]=q2.w;
                a[12]=q3.x; a[13]=q3.y; a[14]=q3.z; a[15]=q3.w;
            }

            float logabs = 0.0f;
            int   neg    = 0;

            #pragma unroll
            for (int k = 0; k < 16; ++k) {
                // partial-pivot search over rows >= k (butterfly within half)
                float val = (i >= k) ? fabsf(a[k]) : -1.0f;
                int   idx = i;
                #pragma unroll
                for (int d = 8; d >= 1; d >>= 1) {
                    float ov = __shfl_xor(val, d, 32);
                    int   oi = __shfl_xor(idx, d, 32);
                    if (ov > val) { val = ov; idx = oi; }
                }
                const int p = idx;  // uniform within half

                // unconditional swap-permute of rows k and p (identity if p==k)
                int src = (p == k) ? i : ((i == k) ? p : ((i == p) ? k : i));
                #pragma unroll
                for (int j = k; j < 16; ++j)
                    a[j] = __shfl(a[j], hbase + src, 32);
                neg ^= (p != k) ? 1 : 0;

                float piv = __shfl(a[k], hbase + k, 32);
                logabs += logf(fabsf(piv));
                neg ^= (piv < 0.0f) ? 1 : 0;

                // eliminate rows below pivot
                float f = (i > k) ? (a[k] / piv) : 0.0f;
                #pragma unroll
                for (int j = k + 1; j < 16; ++j) {
                    float rk = __shfl(a[j], hbase + k, 32);
                    a[j] = fmaf(-f, rk, a[j]);
                }
            }

            float Jq = ldsJ[waveInBlock * 16 + 2 * t + hi];
            if (i == 0) {
                int q = rowBase + 2 * t + hi;
                out[2 * q]     = logabs + Jq;            // Re: logabs + Jastrow
                out[2 * q + 1] = neg ? PI_F : 0.0f;      // Im: log(sign)
            }
        }
    }
}

extern "C" void kernel_launch(void* const* d_in, const int* in_sizes, int n_in,
                              void* d_out, int out_size, void* d_ws, size_t ws_size,
                              hipStream_t stream) {
    const float* nmat = (const float*)d_in[0];
    const float* Mmat = (const float*)d_in[1];
    const float* Wmat = (const float*)d_in[2];
    const float* bvec = (const float*)d_in[3];
    float* out = (float*)d_out;

    int Bn     = in_sizes[0] / N_ORB;
    int nTiles = Bn / 16;                 // B = 131072 -> 8192 tiles
    int blocks = (nTiles + 3) / 4;        // 4 waves per 128-thread block
    if (blocks < 1) blocks = 1;

    njs_kernel<<<blocks, 128, 0, stream>>>(nmat, Mmat, Wmat, bvec, out, nTiles);
}